// upsample_57312043598151
// MI455X (gfx1250) — compile-verified
//
#include <hip/hip_runtime.h>

// 2x nearest-neighbor upsample:
//   in : [B, C, H, W]  f32   (B=16, C=128, H=128, W=128)
//   out: [B, C, 2H, 2W] f32
//
// Pure streaming kernel: 134 MB in + 537 MB out  =>  ~28.8 us floor @ 23.3 TB/s.
// Key identity: with rowId = (b*C+c)*H + h, the even output row is exactly
// 2*rowId (since (b*C+c)*2H + 2h = 2*((b*C+c)*H + h)) -- no div/mod by H needed.
//
// Each thread: 1x b128 NT load (4 input floats) -> 4x b128 NT stores
// (duplicate along W into rows 2*rowId and 2*rowId+1). Grid-stride loop with
// gfx1250 global_prefetch of next iteration's input line into L2.

// native clang vector type: required by __builtin_nontemporal_load/store
typedef float v4f __attribute__((ext_vector_type(4)));

#define W_IN      128
#define QPR_IN    (W_IN / 4)        // 32 input float4's per input row
#define QPR_IN_SH 5                 // log2(QPR_IN)
#define QPR_OUT   (2 * W_IN / 4)    // 64 output float4's per output row

__global__ __launch_bounds__(256) void upsample2x_nn_f32(
    const v4f* __restrict__ in, v4f* __restrict__ out, int nQuads)
{
    const int stride = (int)(gridDim.x * blockDim.x);
    for (int t = (int)(blockIdx.x * blockDim.x + threadIdx.x); t < nQuads; t += stride) {
        // gfx1250 speculative prefetch of next grid-stride iteration's line
        // into L2 (192 MB): harmless if out of range (dropped on xlat fail).
        __builtin_prefetch(in + t + stride, /*rw=*/0, /*locality=*/0);

        // read-once input: non-temporal b128 load
        v4f v = __builtin_nontemporal_load(in + t);

        const int w4    = t & (QPR_IN - 1);   // input quad within row
        const int rowId = t >> QPR_IN_SH;     // (b*C+c)*H + h

        const v4f lo = {v.x, v.x, v.y, v.y};  // out cols 8w..8w+3
        const v4f hi = {v.z, v.z, v.w, v.w};  // out cols 8w+4..8w+7

        // even output row = 2*rowId, odd = 2*rowId+1, each QPR_OUT quads wide
        const long long e = (long long)(2 * rowId) * QPR_OUT + 2 * w4;

        // streamed write-once output: non-temporal b128 stores (don't thrash L2)
        __builtin_nontemporal_store(lo, out + e);
        __builtin_nontemporal_store(hi, out + e + 1);
        __builtin_nontemporal_store(lo, out + e + QPR_OUT);
        __builtin_nontemporal_store(hi, out + e + QPR_OUT + 1);
    }
}

extern "C" void kernel_launch(void* const* d_in, const int* in_sizes, int n_in,
                              void* d_out, int out_size, void* d_ws, size_t ws_size,
                              hipStream_t stream) {
    (void)n_in; (void)out_size; (void)d_ws; (void)ws_size;

    const v4f* in  = (const v4f*)d_in[0];
    v4f*       out = (v4f*)d_out;

    const int nQuads  = in_sizes[0] / 4;   // 8,388,608 float4's

    // 4096 blocks x 256 threads (8 wave32 per block): >1M threads in flight,
    // 8 grid-stride iterations per thread so the prefetch has real distance.
    const int threads = 256;
    const int blocks  = 4096;

    upsample2x_nn_f32<<<blocks, threads, 0, stream>>>(in, out, nQuads);
}